// RegularizedFMNet_64982855188771
// MI455X (gfx1250) — compile-verified
//
#include <hip/hip_runtime.h>
#include <hip/hip_bf16.h>
#include <math.h>

// ---------------------------------------------------------------------------
// RegularizedFMNet for MI455X (gfx1250, wave32, WMMA f32 16x16x4,
// async global->LDS loads, triple-buffered pipeline)
// ---------------------------------------------------------------------------

typedef __attribute__((ext_vector_type(2))) float v2f;
typedef __attribute__((ext_vector_type(8))) float v8f;

__device__ __forceinline__ v8f wmma_f32x4(v2f a, v2f b, v8f c) {
    return __builtin_amdgcn_wmma_f32_16x16x4_f32(
        false, a, false, b, (short)0, c, false, false);
}

__device__ __forceinline__ float4 ldg4(const float* p) {
    return *(const float4*)p;
}

// Async global->LDS 16B load (tracked by ASYNCcnt). LDS operand is the
// wave-relative LDS byte address = low 32 bits of the generic pointer.
__device__ __forceinline__ void async_ld16(void* lds, const float* g) {
    unsigned int l = (unsigned int)(unsigned long long)lds;
    asm volatile("global_load_async_to_lds_b128 %0, %1, off"
                 :: "v"(l), "v"(g) : "memory");
}
__device__ __forceinline__ void wait_async0() {
#if __has_builtin(__builtin_amdgcn_s_wait_asynccnt)
    __builtin_amdgcn_s_wait_asynccnt(0);
#else
    asm volatile("s_wait_asynccnt 0x0" ::: "memory");
#endif
}
__device__ __forceinline__ void wait_async4() {
#if __has_builtin(__builtin_amdgcn_s_wait_asynccnt)
    __builtin_amdgcn_s_wait_asynccnt(4);
#else
    asm volatile("s_wait_asynccnt 0x4" ::: "memory");
#endif
}

#define LAMBDA_REG 0.001f
#define GAMMA_REG 0.5f
#define SPLITS 16
#define NITER 32
#define TK 16          // k-chunk per LDS stage
#define PADK (TK + 4)  // 20: 16B-aligned rows, conflict-free for 16-lane reads

// ---------------------------------------------------------------------------
// Kernel 1: spectral mask  D12[i,k] = mask(gy[i], gx[k]), D21 swapped
// ---------------------------------------------------------------------------
__global__ __launch_bounds__(256) void mask_kernel(
    const float* __restrict__ ex, const float* __restrict__ ey,
    float* __restrict__ d12, float* __restrict__ d21, int K)
{
    __shared__ float red[256];
    __shared__ float gx[256];
    __shared__ float gy[256];
    __shared__ float sS;
    int t = threadIdx.x;
    float m = -1e30f;
    if (t < K) m = fmaxf(ex[t], ey[t]);
    red[t] = m;
    __syncthreads();
    for (int o = 128; o > 0; o >>= 1) {
        if (t < o) red[t] = fmaxf(red[t], red[t + o]);
        __syncthreads();
    }
    if (t == 0) sS = red[0];
    __syncthreads();
    float s = sS;
    if (t < K) {
        gx[t] = powf(ex[t] / s, GAMMA_REG);
        gy[t] = powf(ey[t] / s, GAMMA_REG);
    }
    __syncthreads();
    int KK = K * K;
    for (int e = t; e < KK; e += 256) {
        int i = e / K, k = e - i * K;
        {
            float g2 = gy[i], g1 = gx[k];
            float q2 = 1.f / (g2 * g2 + 1.f);
            float q1 = 1.f / (g1 * g1 + 1.f);
            float re = g2 * q2 - g1 * q1;
            float im = q2 - q1;
            d12[e] = re * re + im * im;
        }
        {
            float g2 = gx[i], g1 = gy[k];
            float q2 = 1.f / (g2 * g2 + 1.f);
            float q1 = 1.f / (g1 * g1 + 1.f);
            float re = g2 * q2 - g1 * q1;
            float im = q2 - q1;
            d21[e] = re * re + im * im;
        }
    }
}

// ---------------------------------------------------------------------------
// Kernel 2: projection GEMM  P = E[K,V] @ F[V,C], split-K over V
// grid = (C/128, K/128, 2*SPLITS); 256 threads = 8 waves; 128x128 tile/WG,
// 32x64 per wave (8 wmma per k-step). Async global->LDS, triple buffered.
// ---------------------------------------------------------------------------
__global__ __launch_bounds__(256) void proj_gemm_kernel(
    const float* __restrict__ Ex, const float* __restrict__ Ey,
    const float* __restrict__ Fx, const float* __restrict__ Fy,
    float* __restrict__ Apart, float* __restrict__ Bpart,
    int V, int C, int K, int chunksPerSplit, int totChunks)
{
    int bz = blockIdx.z;
    int proj = bz & 1, s = bz >> 1;
    const float* E = proj ? Ey : Ex;
    const float* F = proj ? Fy : Fx;
    float* outP = proj ? Bpart : Apart;

    int m0 = blockIdx.y * 128, n0 = blockIdx.x * 128;
    int tid = threadIdx.x;
    int lane = tid & 31, w = tid >> 5;
    int half = lane >> 4, lr = lane & 15;
    int wm = (w >> 1) * 32;   // 4 M-groups of 32 rows
    int wn = (w & 1) * 64;    // 2 N-groups of 64 cols

    __shared__ __align__(16) float et[3][128][PADK];  // [m][k]
    __shared__ __align__(16) float ft[3][TK][128];    // [k][n]

    v8f acc0[4] = {{}, {}, {}, {}};
    v8f acc1[4] = {{}, {}, {}, {}};

    int c0 = s * chunksPerSplit;
    int c1 = c0 + chunksPerSplit;
    if (c1 > totChunks) c1 = totChunks;
    int cFull = V / TK;              // chunks strictly inside [0, V)
    if (cFull > c1) cFull = c1;

    int erow = tid >> 2, ecg = (tid & 3) * 4;       // E: 2 rows/thread (erow, erow+64)
    int frow = tid >> 4, fc = (tid & 15) * 8;       // F: 2 float4/thread in a row
    const float* Erow0 = E + (size_t)(m0 + erow) * V;
    const float* Erow1 = E + (size_t)(m0 + 64 + erow) * V;
    bool vecOK = ((V & 3) == 0) && ((C & 3) == 0);
    if (!vecOK) cFull = c0;  // scalar fallback handles everything

    auto issue = [&](int ch, int b) {
        int kb = ch * TK;
        async_ld16(&et[b][erow][ecg], Erow0 + kb + ecg);
        async_ld16(&et[b][64 + erow][ecg], Erow1 + kb + ecg);
        const float* fp = F + (size_t)(kb + frow) * C + n0 + fc;
        async_ld16(&ft[b][frow][fc], fp);
        async_ld16(&ft[b][frow][fc + 4], fp + 4);
    };
    auto compute = [&](int b) {
        #pragma unroll
        for (int kk = 0; kk < 4; ++kk) {
            int kc = kk * 4 + 2 * half;
            v2f a0, a1;
            a0[0] = et[b][wm + lr][kc];       a0[1] = et[b][wm + lr][kc + 1];
            a1[0] = et[b][wm + 16 + lr][kc];  a1[1] = et[b][wm + 16 + lr][kc + 1];
            #pragma unroll
            for (int ni = 0; ni < 4; ++ni) {
                v2f bv;
                bv[0] = ft[b][kc][wn + ni * 16 + lr];
                bv[1] = ft[b][kc + 1][wn + ni * 16 + lr];
                acc0[ni] = wmma_f32x4(a0, bv, acc0[ni]);
                acc1[ni] = wmma_f32x4(a1, bv, acc1[ni]);
            }
        }
    };

    int nFull = cFull - c0;
    if (nFull > 0) issue(c0, 0);
    if (nFull > 1) issue(c0 + 1, 1);
    for (int i = 0; i < nFull; ++i) {
        // retire chunk i (in-order): leave chunk i+1's 4 loads in flight
        if (i + 1 < nFull) wait_async4(); else wait_async0();
        __syncthreads();
        // buffer (i+2)%3 was last computed at iter i-1 -> safe to refill now
        if (i + 2 < nFull) issue(c0 + i + 2, (i + 2) % 3);
        compute(i % 3);
    }
    // ragged tail (and full scalar fallback when !vecOK)
    for (int ch = (vecOK ? cFull : c0); ch < c1; ++ch) {
        __syncthreads();
        int kb = ch * TK;
        #pragma unroll
        for (int u = 0; u < 4; ++u) {
            int gk = kb + ecg + u;
            et[0][erow][ecg + u] = (gk < V) ? Erow0[gk] : 0.f;
            et[0][64 + erow][ecg + u] = (gk < V) ? Erow1[gk] : 0.f;
        }
        int gk = kb + frow;
        #pragma unroll
        for (int u = 0; u < 8; ++u)
            ft[0][frow][fc + u] = (gk < V) ? F[(size_t)gk * C + n0 + fc + u] : 0.f;
        __syncthreads();
        compute(0);
    }

    size_t base = (size_t)s * K * C;
    #pragma unroll
    for (int mi = 0; mi < 2; ++mi) {
        #pragma unroll
        for (int ni = 0; ni < 4; ++ni) {
            #pragma unroll
            for (int j = 0; j < 8; ++j) {
                int m = m0 + wm + mi * 16 + j + 8 * half;
                int n = n0 + wn + ni * 16 + lr;
                float val = mi ? acc1[ni][j] : acc0[ni][j];
                outP[base + (size_t)m * C + n] = val;
            }
        }
    }
}

// ---------------------------------------------------------------------------
// Kernel 3: split-K reduction (deterministic fixed order)
// ---------------------------------------------------------------------------
__global__ __launch_bounds__(256) void reduce_split_kernel(
    const float* __restrict__ part, float* __restrict__ out, int n)
{
    int i = blockIdx.x * 256 + threadIdx.x;
    if (i < n) {
        float acc = 0.f;
        #pragma unroll
        for (int s = 0; s < SPLITS; ++s) acc += part[(size_t)s * n + i];
        out[i] = acc;
    }
}

// ---------------------------------------------------------------------------
// Kernel 4: Gram GEMMs  O = X[K,C] @ Y[K,C]^T  (z: AAt, BBt, ABt, BAt)
// ---------------------------------------------------------------------------
__global__ __launch_bounds__(256) void gram_gemm_kernel(
    const float* __restrict__ A, const float* __restrict__ B,
    float* __restrict__ AAt, float* __restrict__ BBt,
    float* __restrict__ ABt, float* __restrict__ BAt, int K, int C)
{
    int bz = blockIdx.z;
    const float* X = A; const float* Y = A; float* O = AAt;
    if (bz == 1) { X = B; Y = B; O = BBt; }
    else if (bz == 2) { X = A; Y = B; O = ABt; }
    else if (bz == 3) { X = B; Y = A; O = BAt; }

    int m0 = blockIdx.y * 64, n0 = blockIdx.x * 64;
    int tid = threadIdx.x;
    int lane = tid & 31, w = tid >> 5;
    int half = lane >> 4, lr = lane & 15;
    int wm = (w >> 1) * 16, wn = (w & 1) * 32;

    __shared__ __align__(16) float xt[2][64][PADK];  // [m][k]
    __shared__ __align__(16) float yt[2][64][PADK];  // [n][k]

    v8f acc0 = {};
    v8f acc1 = {};

    int row = tid >> 2, cg = (tid & 3) * 4;
    const float* Xrow = X + (size_t)(m0 + row) * C;
    const float* Yrow = Y + (size_t)(n0 + row) * C;

    auto compute = [&](int b) {
        #pragma unroll
        for (int kk = 0; kk < 4; ++kk) {
            int kc = kk * 4 + 2 * half;
            v2f a;  a[0]  = xt[b][wm + lr][kc];       a[1]  = xt[b][wm + lr][kc + 1];
            v2f b0; b0[0] = yt[b][wn + lr][kc];       b0[1] = yt[b][wn + lr][kc + 1];
            v2f b1; b1[0] = yt[b][wn + 16 + lr][kc];  b1[1] = yt[b][wn + 16 + lr][kc + 1];
            acc0 = wmma_f32x4(a, b0, acc0);
            acc1 = wmma_f32x4(a, b1, acc1);
        }
    };

    int nCh = C / TK, buf = 0;
    {
        float4 xv = ldg4(Xrow + cg), yv = ldg4(Yrow + cg);
        *(float4*)&xt[0][row][cg] = xv;
        *(float4*)&yt[0][row][cg] = yv;
    }
    for (int ch = 0; ch < nCh; ++ch) {
        __syncthreads();
        float4 xv, yv;
        bool nxt = (ch + 1 < nCh);
        if (nxt) {
            int kb = (ch + 1) * TK;
            xv = ldg4(Xrow + kb + cg);
            yv = ldg4(Yrow + kb + cg);
        }
        compute(buf);
        if (nxt) {
            *(float4*)&xt[buf ^ 1][row][cg] = xv;
            *(float4*)&yt[buf ^ 1][row][cg] = yv;
        }
        buf ^= 1;
    }
    #pragma unroll
    for (int j = 0; j < 8; ++j) {
        int m = m0 + wm + j + 8 * half;
        int n = n0 + wn + lr;
        O[(size_t)m * K + n] = acc0[j];
        O[(size_t)m * K + n + 16] = acc1[j];
    }
}

// ---------------------------------------------------------------------------
// Kernel 5: CG init. Systems stored row-major [system i][component k].
// ---------------------------------------------------------------------------
__global__ __launch_bounds__(256) void cg_init_kernel(
    const float* __restrict__ BAt, const float* __restrict__ ABt,
    float* __restrict__ Rt, float* __restrict__ Pt, float* __restrict__ X,
    float* __restrict__ rr, int K)
{
    __shared__ float red[256];
    int i = blockIdx.x, t = threadIdx.x;
    int batch = i / K, sys = i - batch * K;
    const float* rhs = batch ? ABt : BAt;
    float r = 0.f;
    if (t < K) {
        r = rhs[(size_t)sys * K + t];
        size_t g = (size_t)i * K + t;
        Rt[g] = r;
        Pt[g] = r;
        X[g] = 0.f;
    }
    red[t] = r * r;
    __syncthreads();
    for (int o = 128; o > 0; o >>= 1) {
        if (t < o) red[t] += red[t + o];
        __syncthreads();
    }
    if (t == 0) rr[i] = red[0];
}

// ---------------------------------------------------------------------------
// Kernel 6: APt = Pt @ M  (M symmetric -> B tile loaded from rows of M),
// fused epilogue APt += lambda * D .* Pt.  grid = (K/64, K/64, 2)
// ---------------------------------------------------------------------------
__global__ __launch_bounds__(256) void ap_gemm_kernel(
    const float* __restrict__ AAt, const float* __restrict__ BBt,
    const float* __restrict__ PtAll, const float* __restrict__ d12,
    const float* __restrict__ d21, float* __restrict__ APtAll, int K)
{
    int batch = blockIdx.z;
    size_t KK = (size_t)K * K;
    const float* M = batch ? BBt : AAt;
    const float* P = PtAll + (size_t)batch * KK;
    const float* D = batch ? d21 : d12;
    float* O = APtAll + (size_t)batch * KK;

    int m0 = blockIdx.y * 64, n0 = blockIdx.x * 64;
    int tid = threadIdx.x;
    int lane = tid & 31, w = tid >> 5;
    int half = lane >> 4, lr = lane & 15;
    int wm = (w >> 1) * 16, wn = (w & 1) * 32;

    __shared__ __align__(16) float pt[2][64][PADK];   // [i][j]
    __shared__ __align__(16) float mtl[2][64][PADK];  // [k][j] (M symmetric)

    v8f acc0 = {};
    v8f acc1 = {};

    int row = tid >> 2, cg = (tid & 3) * 4;
    const float* Prow = P + (size_t)(m0 + row) * K;
    const float* Mrow = M + (size_t)(n0 + row) * K;

    auto compute = [&](int b) {
        #pragma unroll
        for (int kk = 0; kk < 4; ++kk) {
            int kc = kk * 4 + 2 * half;
            v2f a;  a[0]  = pt[b][wm + lr][kc];        a[1]  = pt[b][wm + lr][kc + 1];
            v2f b0; b0[0] = mtl[b][wn + lr][kc];       b0[1] = mtl[b][wn + lr][kc + 1];
            v2f b1; b1[0] = mtl[b][wn + 16 + lr][kc];  b1[1] = mtl[b][wn + 16 + lr][kc + 1];
            acc0 = wmma_f32x4(a, b0, acc0);
            acc1 = wmma_f32x4(a, b1, acc1);
        }
    };

    int nCh = K / TK, buf = 0;
    {
        float4 pv = ldg4(Prow + cg), mv = ldg4(Mrow + cg);
        *(float4*)&pt[0][row][cg] = pv;
        *(float4*)&mtl[0][row][cg] = mv;
    }
    for (int ch = 0; ch < nCh; ++ch) {
        __syncthreads();
        float4 pv, mv;
        bool nxt = (ch + 1 < nCh);
        if (nxt) {
            int jb = (ch + 1) * TK;
            pv = ldg4(Prow + jb + cg);
            mv = ldg4(Mrow + jb + cg);
        }
        compute(buf);
        if (nxt) {
            *(float4*)&pt[buf ^ 1][row][cg] = pv;
            *(float4*)&mtl[buf ^ 1][row][cg] = mv;
        }
        buf ^= 1;
    }
    #pragma unroll
    for (int j = 0; j < 8; ++j) {
        int m = m0 + wm + j + 8 * half;  // system index i
        int n = n0 + wn + lr;            // component k
        size_t i0 = (size_t)m * K + n;
        size_t i1 = i0 + 16;
        O[i0] = acc0[j] + LAMBDA_REG * D[i0] * P[i0];
        O[i1] = acc1[j] + LAMBDA_REG * D[i1] * P[i1];
    }
}

// ---------------------------------------------------------------------------
// Kernel 7: CG scalar step per system (block per system, 256 threads)
// ---------------------------------------------------------------------------
__global__ __launch_bounds__(256) void cg_step_kernel(
    float* __restrict__ Pt, const float* __restrict__ APt,
    float* __restrict__ Rt, float* __restrict__ X, float* __restrict__ rr, int K)
{
    __shared__ float red[256];
    __shared__ float sAB[2];
    int i = blockIdx.x, t = threadIdx.x;
    size_t g = (size_t)i * K + t;
    float p = 0.f, ap = 0.f, r = 0.f, x = 0.f;
    if (t < K) { p = Pt[g]; ap = APt[g]; r = Rt[g]; x = X[g]; }
    red[t] = p * ap;
    __syncthreads();
    for (int o = 128; o > 0; o >>= 1) {
        if (t < o) red[t] += red[t + o];
        __syncthreads();
    }
    if (t == 0) {
        float pap = red[0];
        sAB[0] = (pap != 0.f) ? (rr[i] / pap) : 0.f;
    }
    __syncthreads();
    float alpha = sAB[0];
    x += alpha * p;
    r -= alpha * ap;
    red[t] = r * r;
    __syncthreads();
    for (int o = 128; o > 0; o >>= 1) {
        if (t < o) red[t] += red[t + o];
        __syncthreads();
    }
    if (t == 0) {
        float rn = red[0];
        float ro = rr[i];
        sAB[1] = (ro != 0.f) ? (rn / ro) : 0.f;
        rr[i] = rn;
    }
    __syncthreads();
    float beta = sAB[1];
    p = r + beta * p;
    if (t < K) { Pt[g] = p; Rt[g] = r; X[g] = x; }
}

// ---------------------------------------------------------------------------
// Host orchestration
// ---------------------------------------------------------------------------
extern "C" void kernel_launch(void* const* d_in, const int* in_sizes, int n_in,
                              void* d_out, int out_size, void* d_ws, size_t ws_size,
                              hipStream_t stream) {
    const float* feat_x  = (const float*)d_in[0];
    const float* feat_y  = (const float*)d_in[1];
    const float* evals_x = (const float*)d_in[2];
    const float* evals_y = (const float*)d_in[3];
    const float* etx     = (const float*)d_in[4];
    const float* ety     = (const float*)d_in[5];

    int K = in_sizes[2];                              // 256
    int V = (int)((long long)in_sizes[4] / K);        // 50000
    int C = (int)((long long)in_sizes[0] / V);        // 512

    float* out = (float*)d_out;                        // [C12 | C21], 2*K*K
    size_t KK = (size_t)K * K;
    size_t KC = (size_t)K * C;

    float* w     = (float*)d_ws;
    float* d12   = w;                 // KK
    float* d21   = d12 + KK;          // KK
    float* A     = d21 + KK;          // KC
    float* B     = A + KC;            // KC
    float* AAt   = B + KC;            // KK
    float* BBt   = AAt + KK;          // KK
    float* ABt   = BBt + KK;          // KK
    float* BAt   = ABt + KK;          // KK
    float* Rt    = BAt + KK;          // 2*KK
    float* Pt    = Rt + 2 * KK;       // 2*KK
    float* APt   = Pt + 2 * KK;       // 2*KK
    float* rr    = APt + 2 * KK;      // 2*K
    float* Apart = rr + 2 * K;        // SPLITS*KC
    float* Bpart = Apart + (size_t)SPLITS * KC;

    // 1. spectral masks
    mask_kernel<<<1, 256, 0, stream>>>(evals_x, evals_y, d12, d21, K);

    // 2. projections with split-K (128x128 tiles)
    int totChunks = (V + TK - 1) / TK;
    int cps = (totChunks + SPLITS - 1) / SPLITS;
    proj_gemm_kernel<<<dim3(C / 128, K / 128, 2 * SPLITS), 256, 0, stream>>>(
        etx, ety, feat_x, feat_y, Apart, Bpart, V, C, K, cps, totChunks);
    reduce_split_kernel<<<(int)((KC + 255) / 256), 256, 0, stream>>>(Apart, A, (int)KC);
    reduce_split_kernel<<<(int)((KC + 255) / 256), 256, 0, stream>>>(Bpart, B, (int)KC);

    // 3. Gram matrices
    gram_gemm_kernel<<<dim3(K / 64, K / 64, 4), 256, 0, stream>>>(
        A, B, AAt, BBt, ABt, BAt, K, C);

    // 4. blocked CG solves: C12 (M=AAt, rhs=BAt rows, diag=D12),
    //    C21 (M=BBt, rhs=ABt rows, diag=D21). X lives in d_out.
    cg_init_kernel<<<2 * K, 256, 0, stream>>>(BAt, ABt, Rt, Pt, out, rr, K);
    for (int it = 0; it < NITER; ++it) {
        ap_gemm_kernel<<<dim3(K / 64, K / 64, 2), 256, 0, stream>>>(
            AAt, BBt, Pt, d12, d21, APt, K);
        cg_step_kernel<<<2 * K, 256, 0, stream>>>(Pt, APt, Rt, out, rr, K);
    }
}